// VoxelSampler_denet_23605140259601
// MI455X (gfx1250) — compile-verified
//
#include <hip/hip_runtime.h>
#include <stdint.h>

// ---- problem constants (match reference setup_inputs) ----
#define VSC   0.4f      // coarse voxel size (x,y)
#define VSF   0.2f      // fine voxel size (x,y,z)
#define PC0   (-75.2f)  // xy range start
#define PCZ   (-2.0f)   // z range start
#define GXC   376
#define GYC   376
#define GXF   752
#define GYF   752
#define GZF   30
#define KPPV  32        // max points per coarse voxel
#define K2PPV 5         // max points per fine voxel
#define NS    32        // num_sample
#define GAMMAF 1.1f
#define BB 2
#define TT 4
#define RR 128

// fine-voxel hash table
#define HKBITS 17
#define HK  (1u << HKBITS)
#define HKM (HK - 1u)
#define EMPTYK 0xFFFFFFFFu

// TDM fill chunking: each block DMA-stores 8KB to hkey and 8KB to hcnt
#define ZCHUNK_B   8192
#define ZCHUNK_DW  (ZCHUNK_B / 4)
#define ZBLOCKS    ((HK * 4) / ZCHUNK_B)   // 64

typedef unsigned int u32;
typedef unsigned long long u64;
typedef u32 u32x4 __attribute__((ext_vector_type(4)));
typedef u32 u32x8 __attribute__((ext_vector_type(8)));

// ---------------------------------------------------------------------------
// Build a 1-D TDM descriptor (D# groups 0 and 1) per CDNA5 ISA
// §8.3/§8.4: group0 = {count=1, lds_addr, global_addr[56:0], type=2},
// group1 = {wg_mask=0, data_size=4B, tile_dim0=ndw, tensor_dim0=ndw,
//           tensor_dim1=1, tensor_dim0_stride=ndw}.
// ---------------------------------------------------------------------------
__device__ __forceinline__ void tdm_desc_1d(u32 ldsOff, u64 gaddr, u32 ndw,
                                            u32x4* g0, u32x8* g1) {
  u32x4 a;
  a.x = 1u;                                              // count=1, user mode
  a.y = ldsOff;                                          // lds_addr (bytes)
  a.z = (u32)gaddr;                                      // global_addr[31:0]
  a.w = ((u32)(gaddr >> 32) & 0x1FFFFFFu) | (2u << 30);  // ga[56:32] | type=2
  u32x8 b;
  b.s0 = 2u << 16;                        // wg_mask=0, data_size=2 (4 bytes)
  b.s1 = (ndw & 0xFFFFu) << 16;           // tensor_dim0[15:0] (bits 79:48)
  b.s2 = (ndw >> 16) | (1u << 16);        // tensor_dim0[31:16] | tensor_dim1=1
  b.s3 = (ndw & 0xFFFFu) << 16;           // tensor_dim1 hi=0 | tile_dim0=ndw
  b.s4 = 0u;                              // tile_dim1=0 (1-D), tile_dim2=0
  b.s5 = ndw;                             // tensor_dim0_stride[31:0]
  b.s6 = 0u;                              // stride hi, dim1_stride lo
  b.s7 = 0u;
  *g0 = a; *g1 = b;
}

// ---------------------------------------------------------------------------
// Kernel 0: per-box derived parameters (runs once for all (bs,t,r))
// params[idx*10]: bx, by, zlim, cur_r^2, rad, qx, qy, q1x, q1y, pad
// ---------------------------------------------------------------------------
__global__ void k_boxparams(const float* __restrict__ traj,
                            const float* __restrict__ back,
                            const unsigned char* __restrict__ vlen,
                            float* __restrict__ params) {
  int idx = blockIdx.x * blockDim.x + threadIdx.x;
  if (idx >= BB * TT * RR) return;
  const float* tb = traj + (size_t)idx * 7;
  const float* bk = back + (size_t)idx * 7;
  bool v = vlen[idx] != 0;
  float bx = v ? 0.5f * (tb[0] + bk[0]) : bk[0];
  float by = v ? 0.5f * (tb[1] + bk[1]) : bk[1];
  float bz = bk[2], dx = bk[3], dy = bk[4], dz = bk[5];
  float cur_r = sqrtf(dx * dx + dy * dy) * 0.5f * GAMMAF;
  float rad = ceilf(cur_r / VSC);
  float* p = params + (size_t)idx * 10;
  p[0] = bx; p[1] = by;
  p[2] = bz + 0.6f * dz;          // z limit
  p[3] = cur_r * cur_r;           // radius^2
  p[4] = rad;                     // coarse radius in cells
  p[5] = floorf((bx - PC0) / VSC);
  p[6] = floorf((by - PC0) / VSC);
  p[7] = floorf((bx - PC0) / VSF);
  p[8] = floorf((by - PC0) / VSF);
  p[9] = 0.0f;
}

// ---------------------------------------------------------------------------
// Kernel Z1: zero coarse counts + coarse mask (small: ~0.7 MB)
// ---------------------------------------------------------------------------
__global__ void k_zero(unsigned* __restrict__ ccnt,
                       unsigned char* __restrict__ cmask) {
  int i = blockIdx.x * blockDim.x + threadIdx.x;
  int stride = gridDim.x * blockDim.x;
  for (int j = i; j < GXC * GYC; j += stride) { ccnt[j] = 0u; cmask[j] = 0; }
}

// ---------------------------------------------------------------------------
// Kernel Z2: re-init the 1 MB hash table (hkey=0xFF.., hcnt=0) via the
// Tensor Data Mover. Each block fills a 16 KB LDS staging buffer once and
// issues two TENSOR_STORE_FROM_LDS DMA ops (8 KB each), completing on
// TENSORcnt. This is bulk pattern-fill — exactly what the TDM is for.
// ---------------------------------------------------------------------------
__global__ __launch_bounds__(256) void k_zerohash_tdm(unsigned* __restrict__ hkey,
                                                      unsigned* __restrict__ hcnt) {
  __shared__ u32 zbuf[2 * ZCHUNK_DW];   // [0..2047]=0xFFFFFFFF, [2048..4095]=0
  for (int j = threadIdx.x; j < ZCHUNK_DW; j += blockDim.x) {
    zbuf[j] = EMPTYK;
    zbuf[ZCHUNK_DW + j] = 0u;
  }
  __syncthreads();                       // LDS writes visible before DMA reads
  if (threadIdx.x < 32) {                // wave 0 issues the tensor DMAs
    u32 ldsFF = (u32)(size_t)(void*)&zbuf[0];
    u32 ldsZ  = (u32)(size_t)(void*)&zbuf[ZCHUNK_DW];
    u64 gk = (u64)(size_t)(hkey + (size_t)blockIdx.x * ZCHUNK_DW);
    u64 gc = (u64)(size_t)(hcnt + (size_t)blockIdx.x * ZCHUNK_DW);
    u32x4 g0a, g0b; u32x8 g1a, g1b;
    tdm_desc_1d(ldsFF, gk, ZCHUNK_DW, &g0a, &g1a);
    tdm_desc_1d(ldsZ,  gc, ZCHUNK_DW, &g0b, &g1b);
    asm volatile("tensor_store_from_lds %0, %1" :: "s"(g0a), "s"(g1a) : "memory");
    asm volatile("tensor_store_from_lds %0, %1" :: "s"(g0b), "s"(g1b) : "memory");
    __builtin_amdgcn_s_wait_tensorcnt(0);
  }
}

// ---------------------------------------------------------------------------
// Kernel M: mark coarse cells within |q - c| < rad of any box (per combo)
// ---------------------------------------------------------------------------
__global__ void k_mask(const float* __restrict__ params,
                       unsigned char* __restrict__ cmask, int bs, int t) {
  int r = blockIdx.x;
  const float* p = params + (size_t)((bs * TT + t) * RR + r) * 10;
  int qx = (int)p[5], qy = (int)p[6], ri = (int)p[4];
  int x0 = max(qx - ri + 1, 0), x1 = min(qx + ri - 1, GXC - 1);
  int y0 = max(qy - ri + 1, 0), y1 = min(qy + ri - 1, GYC - 1);
  if (x1 < x0 || y1 < y0) return;
  int wx = x1 - x0 + 1, wy = y1 - y0 + 1, S = wx * wy;
  for (int i = threadIdx.x; i < S; i += blockDim.x) {
    int cy = y0 + i / wx, cx = x0 + i % wx;
    cmask[cy * GXC + cx] = 1;  // idempotent racy store: fine
  }
}

// ---------------------------------------------------------------------------
// Kernel P: per-point pass (per combo). Async-stages each point's 24 bytes
// into LDS with 3x global_load_async_to_lds_b64 (ASYNCcnt path), then
// filters by (bs,t), coarse rank via atomic, near-mask check, and inserts
// survivors into the fine-voxel hash table.
// ---------------------------------------------------------------------------
__global__ __launch_bounds__(256) void k_points(
    const float* __restrict__ pts, int N,
    const unsigned char* __restrict__ cmask,
    unsigned* __restrict__ ccnt,
    unsigned* __restrict__ hkey, unsigned* __restrict__ hcnt,
    float* __restrict__ hpts, int bs, int t) {
  __shared__ float sbuf[256 * 6];
  int tid = threadIdx.x;
  int p = blockIdx.x * blockDim.x + tid;

  unsigned ldsOff = (unsigned)(size_t)(void*)&sbuf[tid * 6];  // addr[31:0] = LDS offset
  if (p < N) {
    u64 ga = (u64)(size_t)(pts + (size_t)p * 6);   // 24B per point, 8-aligned
#pragma unroll
    for (int j = 0; j < 3; j++) {
      asm volatile("global_load_async_to_lds_b64 %0, %1, off"
                   :: "v"(ldsOff + 8u * (unsigned)j),
                      "v"(ga + 8ull * (u64)j)
                   : "memory");
    }
  }
  asm volatile("s_wait_asynccnt 0" ::: "memory");
  if (p >= N) return;

  float b  = sbuf[tid * 6 + 0];
  float px = sbuf[tid * 6 + 1];
  float py = sbuf[tid * 6 + 2];
  float pz = sbuf[tid * 6 + 3];
  float pi = sbuf[tid * 6 + 4];
  float pt = sbuf[tid * 6 + 5];

  if (b != (float)bs) return;
  if (fabsf(pt - 0.1f * (float)t) >= 0.001f) return;

  int cx = (int)floorf((px - PC0) / VSC);
  int cy = (int)floorf((py - PC0) / VSC);
  int cz = (int)floorf((pz - PCZ) / 6.0f);   // gz == 1
  if (cx < 0 || cx >= GXC || cy < 0 || cy >= GYC || cz != 0) return;

  int cell = cy * GXC + cx;
  unsigned rank = atomicAdd(&ccnt[cell], 1u);
  if (rank >= KPPV) return;          // only first K points per coarse voxel survive
  if (!cmask[cell]) return;          // voxel not near any box

  // fine voxel insert
  int fx = (int)floorf((px - PC0) / VSF);
  int fy = (int)floorf((py - PC0) / VSF);
  int fz = (int)floorf((pz - PCZ) / VSF);
  if (fx < 0 || fx >= GXF || fy < 0 || fy >= GYF || fz < 0 || fz >= GZF) return;
  unsigned vid = ((unsigned)fz * GYF + (unsigned)fy) * GXF + (unsigned)fx;
  unsigned h = (vid * 2654435761u) >> (32 - HKBITS);
  int slot = -1;
  for (int pr = 0; pr < 128; pr++) {
    unsigned prev = atomicCAS(&hkey[h], EMPTYK, vid);
    if (prev == EMPTYK || prev == vid) { slot = (int)h; break; }
    h = (h + 1u) & HKM;
  }
  if (slot < 0) return;
  unsigned c = atomicAdd(&hcnt[slot], 1u);
  if (c < K2PPV) {
    float* dst = hpts + (size_t)slot * 25 + (size_t)c * 5;
    dst[0] = px; dst[1] = py; dst[2] = pz; dst[3] = pi; dst[4] = pt;
  }
}

// ---------------------------------------------------------------------------
// Kernel S: per-box sampling (per combo). One wave32 per box.
// Phase 1: scan fine-grid window in voxel-id order, ballot-compact first 32
//          occupied voxels. Hash probes hit L2 (table ~15 MB << 192 MB L2);
//          global_prefetch hides the hcnt line behind the hkey load.
// Phase 2: ballot-ordered selection of first 32 points passing the
//          cylinder/z test; zero-fill the remainder.
// ---------------------------------------------------------------------------
__global__ __launch_bounds__(32) void k_sample(
    const float* __restrict__ params,
    const unsigned* __restrict__ hkey, const unsigned* __restrict__ hcnt,
    const float* __restrict__ hpts, float* __restrict__ out, int bs, int t) {
  int r = blockIdx.x;
  int lane = threadIdx.x;
  const float* p = params + (size_t)((bs * TT + t) * RR + r) * 10;
  float bx = p[0], by = p[1], zlim = p[2], cr2 = p[3];
  int ri = (int)p[4];
  int q1x = (int)p[7], q1y = (int)p[8];
  int rad2 = 2 * ri;
  int x0 = max(q1x - rad2 + 1, 0), x1 = min(q1x + rad2 - 1, GXF - 1);
  int y0 = max(q1y - rad2 + 1, 0), y1 = min(q1y + rad2 - 1, GYF - 1);

  __shared__ int vlist[NS];
  __shared__ int vcnt[NS];
  unsigned ltmask = (lane == 31) ? 0x7FFFFFFFu : ((1u << lane) - 1u);

  int total = 0;
  if (x1 >= x0 && y1 >= y0) {
    int wx = x1 - x0 + 1, wy = y1 - y0 + 1;
    int layer = wx * wy;
    int S = GZF * layer;
    for (int base = 0; base < S && total < NS; base += 32) {
      int i = base + lane;
      bool occ = false; int hidx = -1; int cnt = 0;
      if (i < S) {
        int cz = i / layer, rem = i - cz * layer;
        int cy = y0 + rem / wx, cx = x0 + rem % wx;
        unsigned vid = ((unsigned)cz * GYF + (unsigned)cy) * GXF + (unsigned)cx;
        unsigned h = (vid * 2654435761u) >> (32 - HKBITS);
        for (int pr = 0; pr < 128; pr++) {
          __builtin_prefetch((const void*)&hcnt[h], 0, 1);  // global_prefetch
          unsigned k = hkey[h];
          if (k == vid) {
            unsigned c = hcnt[h];
            if (c > 0u) { occ = true; hidx = (int)h; cnt = (int)min(c, (unsigned)K2PPV); }
            break;
          }
          if (k == EMPTYK) break;
          h = (h + 1u) & HKM;
        }
      }
      unsigned bal = (unsigned)__ballot(occ);
      int pos = total + __popc(bal & ltmask);
      if (occ && pos < NS) { vlist[pos] = hidx; vcnt[pos] = cnt; }
      total += __popc(bal);
    }
  }
  __syncthreads();
  int found = min(total, NS);

  float* ob = out + (size_t)(((bs * RR + r) * TT + t) * NS) * 5;
  int outc = 0;
  for (int k2 = 0; k2 < K2PPV && outc < NS; k2++) {
    float v0 = 0.f, v1 = 0.f, v2 = 0.f, v3 = 0.f, v4 = 0.f;
    if (lane < found && k2 < vcnt[lane]) {
      const float* sp = hpts + (size_t)vlist[lane] * 25 + (size_t)k2 * 5;
      v0 = sp[0]; v1 = sp[1]; v2 = sp[2]; v3 = sp[3]; v4 = sp[4];
    }
    float ddx = v0 - bx, ddy = v1 - by;
    bool pass = (ddx * ddx + ddy * ddy < cr2) && (v2 <= zlim);
    unsigned bal = (unsigned)__ballot(pass);
    int pos = outc + __popc(bal & ltmask);
    if (pass && pos < NS) {
      float* o = ob + (size_t)pos * 5;
      o[0] = v0; o[1] = v1; o[2] = v2; o[3] = v3; o[4] = v4;
    }
    outc += __popc(bal);
  }
  int fc = min(outc, NS);
  for (int row = fc + lane; row < NS; row += 32) {
    float* o = ob + (size_t)row * 5;
    o[0] = 0.f; o[1] = 0.f; o[2] = 0.f; o[3] = 0.f; o[4] = 0.f;
  }
}

// ---------------------------------------------------------------------------
// host launcher
// ---------------------------------------------------------------------------
extern "C" void kernel_launch(void* const* d_in, const int* in_sizes, int n_in,
                              void* d_out, int out_size, void* d_ws, size_t ws_size,
                              hipStream_t stream) {
  (void)n_in; (void)out_size; (void)ws_size;
  const float* pts  = (const float*)d_in[0];
  const float* traj = (const float*)d_in[1];
  const float* back = (const float*)d_in[2];
  const unsigned char* vlen = (const unsigned char*)d_in[3];
  int N = in_sizes[0] / 6;
  float* out = (float*)d_out;

  // workspace carve (~15 MB, fully L2-resident on MI455X's 192 MB L2)
  char* ws = (char*)d_ws;
  size_t off = 0;
  auto carve = [&](size_t bytes) -> void* {
    void* ptr = (void*)(ws + off);
    off += (bytes + 255) & ~(size_t)255;
    return ptr;
  };
  float*         params = (float*)carve((size_t)BB * TT * RR * 10 * sizeof(float));
  unsigned*      ccnt   = (unsigned*)carve((size_t)GXC * GYC * sizeof(unsigned));
  unsigned char* cmask  = (unsigned char*)carve((size_t)GXC * GYC);
  unsigned*      hkey   = (unsigned*)carve((size_t)HK * sizeof(unsigned));
  unsigned*      hcnt   = (unsigned*)carve((size_t)HK * sizeof(unsigned));
  float*         hpts   = (float*)carve((size_t)HK * 25 * sizeof(float));

  k_boxparams<<<(BB * TT * RR + 255) / 256, 256, 0, stream>>>(traj, back, vlen, params);

  for (int bs = 0; bs < BB; bs++) {
    for (int t = 0; t < TT; t++) {
      k_zero<<<256, 256, 0, stream>>>(ccnt, cmask);
      k_zerohash_tdm<<<ZBLOCKS, 256, 0, stream>>>(hkey, hcnt);
      k_mask<<<RR, 256, 0, stream>>>(params, cmask, bs, t);
      k_points<<<(N + 255) / 256, 256, 0, stream>>>(pts, N, cmask, ccnt, hkey, hcnt, hpts, bs, t);
      k_sample<<<RR, 32, 0, stream>>>(params, hkey, hcnt, hpts, out, bs, t);
    }
  }
}